// GCNConv_30081950941518
// MI455X (gfx1250) — compile-verified
//
#include <hip/hip_runtime.h>
#include <stdint.h>

// ---------------------------------------------------------------------------
// GCNConv: out = adj @ (x @ weight)   (reassociated: (adj@x)@W == adj@(x@W))
//   adj: [16384,16384] fp32 (1 GiB, read once -> HBM bound, ~46us @ 23.3TB/s)
//   x  : [16384,64] fp32, W: [64,64] fp32, xw staged in d_ws (4 MiB, L2-resident)
// Kernel 1: xw = x @ W   (tiny, VALU)
// Kernel 2: out = adj @ xw  via V_WMMA_F32_16X16X4_F32, tiles staged into LDS
//           by the Tensor Data Mover (double-buffered, padded for bank-free ds)
// ---------------------------------------------------------------------------

typedef __attribute__((ext_vector_type(2))) float v2f;
typedef __attribute__((ext_vector_type(8))) float v8f;
typedef __attribute__((ext_vector_type(4))) unsigned int u32x4;
typedef __attribute__((ext_vector_type(4))) int i32x4;
typedef __attribute__((ext_vector_type(8))) int i32x8;

#define N_NODES 16384
#define C 64
#define MB 32                     // rows of adj per workgroup (2 m-tiles)
#define KB 64                     // K chunk
#define NCHUNK (N_NODES / KB)     // 256
#define ADJ_STRIDE 68             // KB+4: A-frag ds_load_b64 -> banks 4m+{0..3}, conflict-free
#define XW_STRIDE 72              // 2*S%64==16: hi/lo half-wave b32 reads hit disjoint banks
#define A0 0
#define A1 (MB * ADJ_STRIDE)                  // 2176
#define X0 (2 * MB * ADJ_STRIDE)              // 4352
#define X1 (X0 + KB * XW_STRIDE)              // 8960
#define SMEM_FLOATS (X0 + 2 * KB * XW_STRIDE) // 13568 floats = 54,272 B

#if __has_builtin(__builtin_amdgcn_tensor_load_to_lds) && __has_builtin(__builtin_amdgcn_s_wait_tensorcnt)
#define HAVE_TDM 1
#else
#define HAVE_TDM 0
#endif

// Escape hatch: the TDM engine writes LDS behind the compiler's back (the LDS
// address goes into the descriptor as an integer, so the shared array never
// "escapes" in IR and LLVM folds loads from it to undef). Feeding the pointer
// into an opaque asm with a memory clobber forces the compiler to treat the
// array as externally written, keeping the ds_load fragments real.
__device__ __forceinline__ void lds_publish(const void* p) {
  asm volatile("" : : "v"(p) : "memory");
}

#if HAVE_TDM
// 2D TDM load: tile_d0 x tile_d1 elements (4B each) from a row-major tensor with
// row stride stride0, into LDS at lds_byte, padding (pad_amount_code+1) dwords
// after every 64 dwords stored (pad_interval code 5) -> padded LDS strides.
__device__ __forceinline__ void tdm_load_2d(unsigned lds_byte, unsigned long long gaddr,
                                            unsigned tensor_d0, unsigned tensor_d1,
                                            unsigned tile_d0, unsigned tile_d1,
                                            unsigned stride0, unsigned pad_amount_code) {
  u32x4 g0;
  g0.x = 1u;                                     // count=1 (one valid descriptor)
  g0.y = lds_byte;                               // LDS byte address
  g0.z = (unsigned)(gaddr & 0xffffffffull);      // global_addr[31:0]
  g0.w = (unsigned)(gaddr >> 32) | (2u << 30);   // global_addr[56:32] | type=2 ("image")

  i32x8 g1;
  // data_size=2 (4B) | pad_enable | pad_interval=5 (64 dwords) | pad_amount code
  g1[0] = (int)((2u << 16) | (1u << 20) | (5u << 22) | (pad_amount_code << 25));
  g1[1] = (int)((tensor_d0 & 0xffffu) << 16);                        // tensor_dim0[15:0]
  g1[2] = (int)((tensor_d0 >> 16) | ((tensor_d1 & 0xffffu) << 16));  // dim0[31:16]|dim1[15:0]
  g1[3] = (int)((tensor_d1 >> 16) | (tile_d0 << 16));                // dim1[31:16]|tile_dim0
  g1[4] = (int)tile_d1;                                              // tile_dim1 (tile_dim2=0)
  g1[5] = (int)stride0;                                              // tensor_dim0_stride[31:0]
  g1[6] = 0;
  g1[7] = 0;

  i32x4 z4 = {0, 0, 0, 0};
#if __clang_major__ >= 23
  i32x8 z8 = {0, 0, 0, 0, 0, 0, 0, 0};
  __builtin_amdgcn_tensor_load_to_lds(g0, g1, z4, z4, z8, 0);
#else
  __builtin_amdgcn_tensor_load_to_lds(g0, g1, z4, z4, 0);
#endif
}
#endif // HAVE_TDM

// ---------------------------------------------------------------------------
// Kernel 1: xw[i][j] = sum_k x[i][k] * W[k][j]    (0.4% of total FLOPs)
// ---------------------------------------------------------------------------
__global__ __launch_bounds__(256) void gcn_xw_kernel(const float* __restrict__ x,
                                                     const float* __restrict__ w,
                                                     float* __restrict__ xw) {
  __shared__ float ws[C * C];
  const unsigned tid = threadIdx.x;
#pragma unroll
  for (int i = 0; i < (C * C) / 256; ++i) ws[tid + 256u * i] = w[tid + 256u * i];
  __syncthreads();

  const size_t row = (size_t)blockIdx.x * 256 + tid;
  const float* xr = x + row * C;
  float acc[C];
#pragma unroll
  for (int j = 0; j < C; ++j) acc[j] = 0.0f;
  for (int k = 0; k < C; ++k) {
    const float xv = xr[k];
#pragma unroll
    for (int j = 0; j < C; ++j) acc[j] += xv * ws[k * C + j];  // ws read is lane-uniform
  }
  float* o = xw + row * C;
#pragma unroll
  for (int j = 0; j < C; ++j) o[j] = acc[j];
}

// ---------------------------------------------------------------------------
// Kernel 2: out = adj @ xw.  256 threads = 8 waves; wave (mtile,ntile) owns a
// 16x16 f32 accumulator; K=16384 swept in KB=64 chunks, double-buffered in LDS.
// ---------------------------------------------------------------------------
__global__ __launch_bounds__(256) void gcn_adj_wmma_kernel(const float* __restrict__ adj,
                                                           const float* __restrict__ xw,
                                                           float* __restrict__ out) {
  __shared__ float smem[SMEM_FLOATS] __attribute__((aligned(16)));

  const unsigned tid = threadIdx.x;
  const unsigned lane = tid & 31u;
  const unsigned laneN = lane & 15u;
  const unsigned hi = lane >> 4;           // A/B fragment half-wave (K split)
  const unsigned wv = tid >> 5;            // 0..7
  const unsigned mtile = wv >> 2;          // 0..1
  const unsigned ntile = wv & 3u;          // 0..3
  const unsigned row0 = blockIdx.x * MB;

  v8f acc = {0, 0, 0, 0, 0, 0, 0, 0};

#if HAVE_TDM
  // Scalar-uniform wave id so the branch is an s_cbranch (TDM ignores EXEC,
  // so exec-masking would NOT stop waves 1..7 from issuing redundant TDM ops).
  const bool wave0 = ((__builtin_amdgcn_readfirstlane((int)tid) >> 5) == 0);

  if (wave0) {
    tdm_load_2d(A0 * 4, (unsigned long long)(uintptr_t)(adj + (size_t)row0 * N_NODES),
                N_NODES, N_NODES, KB, MB, N_NODES, /*pad 4 dw*/ 3);
    tdm_load_2d(X0 * 4, (unsigned long long)(uintptr_t)xw,
                C, N_NODES, KB, KB, C, /*pad 8 dw*/ 7);
  }

  for (int c = 0; c < NCHUNK; ++c) {
    if (wave0) __builtin_amdgcn_s_wait_tensorcnt(0);  // chunk c resident in LDS
    __syncthreads();                                   // publish to all waves
    lds_publish(smem);                                 // TDM wrote LDS: keep ds loads real
    const int buf = c & 1;
    if (wave0 && (c + 1 < NCHUNK)) {                   // prefetch next chunk
      const int nb = (c + 1) & 1;
      tdm_load_2d((nb ? A1 : A0) * 4,
                  (unsigned long long)(uintptr_t)(adj + (size_t)row0 * N_NODES + (size_t)(c + 1) * KB),
                  N_NODES, N_NODES, KB, MB, N_NODES, 3);
      tdm_load_2d((nb ? X1 : X0) * 4,
                  (unsigned long long)(uintptr_t)(xw + (size_t)(c + 1) * KB * C),
                  C, N_NODES, KB, KB, C, 7);
    }
    const float* aB = smem + (buf ? A1 : A0);
    const float* xB = smem + (buf ? X1 : X0);
    const float* ap = aB + (mtile * 16 + laneN) * ADJ_STRIDE + 2 * hi;
    const float* xp = xB + (2 * hi) * XW_STRIDE + ntile * 16 + laneN;
#pragma unroll
    for (int k = 0; k < KB; k += 4) {
      v2f a = *(const v2f*)(ap + k);                   // adj[m][k+2hi .. +1]
      v2f b;
      b.x = xp[(unsigned)k * XW_STRIDE];               // xw[k+2hi  ][n]
      b.y = xp[(unsigned)k * XW_STRIDE + XW_STRIDE];   // xw[k+2hi+1][n]
      acc = __builtin_amdgcn_wmma_f32_16x16x4_f32(false, a, false, b, (short)0, acc,
                                                  false, false);
    }
  }
#else
  // Fallback: synchronous cooperative loads into the same padded LDS layout.
  for (int c = 0; c < NCHUNK; ++c) {
    __syncthreads();
    {
      const unsigned idx = tid * 8u;                   // adj tile: 2048 floats
      const unsigned r = idx >> 6, col = idx & 63u;
      const float4* src = (const float4*)(adj + (size_t)(row0 + r) * N_NODES + (size_t)c * KB + col);
      *(float4*)(smem + A0 + r * ADJ_STRIDE + col) = src[0];
      *(float4*)(smem + A0 + r * ADJ_STRIDE + col + 4) = src[1];
    }
    {
      const unsigned idx = tid * 16u;                  // xw tile: 4096 floats
      const unsigned r = idx >> 6, col = idx & 63u;
      const float4* src = (const float4*)(xw + (size_t)c * KB * C + (size_t)r * C + col);
#pragma unroll
      for (int q = 0; q < 4; ++q)
        *(float4*)(smem + X0 + r * XW_STRIDE + col + 4 * q) = src[q];
    }
    __syncthreads();
    const float* ap = smem + A0 + (mtile * 16 + laneN) * ADJ_STRIDE + 2 * hi;
    const float* xp = smem + X0 + (2 * hi) * XW_STRIDE + ntile * 16 + laneN;
#pragma unroll
    for (int k = 0; k < KB; k += 4) {
      v2f a = *(const v2f*)(ap + k);
      v2f b;
      b.x = xp[(unsigned)k * XW_STRIDE];
      b.y = xp[(unsigned)k * XW_STRIDE + XW_STRIDE];
      acc = __builtin_amdgcn_wmma_f32_16x16x4_f32(false, a, false, b, (short)0, acc,
                                                  false, false);
    }
  }
#endif

  // D layout (32-bit 16x16): VGPR r -> lanes 0-15: M=r, N=lane; lanes 16-31: M=8+r.
  float* o = out + (size_t)(row0 + mtile * 16 + hi * 8) * C + ntile * 16 + laneN;
#pragma unroll
  for (int r = 0; r < 8; ++r) o[(size_t)r * C] = acc[r];
}

extern "C" void kernel_launch(void* const* d_in, const int* in_sizes, int n_in,
                              void* d_out, int out_size, void* d_ws, size_t ws_size,
                              hipStream_t stream) {
  const float* x = (const float*)d_in[0];     // [16384, 64]
  const float* adj = (const float*)d_in[1];   // [16384, 16384]
  const float* w = (const float*)d_in[2];     // [64, 64]
  float* out = (float*)d_out;                 // [16384, 64]
  float* xw = (float*)d_ws;                   // scratch: 16384*64*4 = 4 MiB

  gcn_xw_kernel<<<dim3(N_NODES / 256), dim3(256), 0, stream>>>(x, w, xw);
  gcn_adj_wmma_kernel<<<dim3(N_NODES / MB), dim3(256), 0, stream>>>(adj, xw, out);
}